// GCN_54838142435789
// MI455X (gfx1250) — compile-verified
//
#include <hip/hip_runtime.h>
#include <math.h>

typedef __attribute__((ext_vector_type(2))) float v2f;
typedef __attribute__((ext_vector_type(8))) float v8f;

#define NF 64  // feature width == hidden width == 64

// ---------------- degree / norm ----------------
__global__ void k_init_deg(float* __restrict__ deg, int n) {
    int i = blockIdx.x * blockDim.x + threadIdx.x;
    if (i < n) deg[i] = 1.0f;  // self-loop contribution
}

__global__ void k_deg_accum(const int* __restrict__ dst, float* __restrict__ deg, int e) {
    int i = blockIdx.x * blockDim.x + threadIdx.x;
    if (i < e) unsafeAtomicAdd(&deg[dst[i]], 1.0f);
}

__global__ void k_dinv(float* __restrict__ deg, int n) {
    int i = blockIdx.x * blockDim.x + threadIdx.x;
    if (i < n) deg[i] = rsqrtf(deg[i]);  // deg >= 1 always (self loop)
}

// ---------------- WMMA GEMM: Out[N,64] = (relu?)A[N,64] @ W[64,64] ----------------
// 128 threads = 4 waves per block; each wave computes a 16x64 output tile.
// W is staged in LDS pre-paired as float2 so each B fragment is one ds_load_b64.
__global__ __launch_bounds__(128) void k_gemm64(const float* __restrict__ A,
                                                const float* __restrict__ W,
                                                float* __restrict__ Out,
                                                int nrows, int relu_in) {
    // sW[(kb*2 + half)*64 + n] = { W[4*kb + 2*half][n], W[4*kb + 2*half + 1][n] }
    __shared__ v2f sW[32 * NF];  // 16 KB
    int tid = threadIdx.x;
    for (int i = tid; i < 32 * NF; i += 128) {
        int n   = i & 63;
        int kb2 = i >> 6;                       // kb*2 + half, 0..31
        int k   = ((kb2 >> 1) << 2) + ((kb2 & 1) << 1);
        v2f w;
        w.x = W[k * NF + n];
        w.y = W[(k + 1) * NF + n];
        sW[i] = w;
    }
    __syncthreads();

    int wave = tid >> 5;
    int lane = tid & 31;
    int row_base = (blockIdx.x * 4 + wave) * 16;
    if (row_base >= nrows) return;  // whole-wave uniform: EXEC stays all-ones

    int m    = lane & 15;   // A row within tile / output column within N-tile
    int half = lane >> 4;   // K sub-pair select

    // Preload all 16 A fragments (pipelined b64 loads), then compute from LDS only.
    const float* Arow = A + (long long)(row_base + m) * NF + (half << 1);
    v2f afrag[16];
#pragma unroll
    for (int kb = 0; kb < 16; ++kb)
        afrag[kb] = *(const v2f*)(Arow + (kb << 2));
    if (relu_in) {
#pragma unroll
        for (int kb = 0; kb < 16; ++kb) {
            afrag[kb].x = fmaxf(afrag[kb].x, 0.0f);
            afrag[kb].y = fmaxf(afrag[kb].y, 0.0f);
        }
    }

    v8f acc0 = {}, acc1 = {}, acc2 = {}, acc3 = {};
#pragma unroll
    for (int kb = 0; kb < 16; ++kb) {
        const v2f* bb = &sW[((kb << 1) + half) * NF + m];
        v2f b0 = bb[0], b1 = bb[16], b2 = bb[32], b3 = bb[48];
        acc0 = __builtin_amdgcn_wmma_f32_16x16x4_f32(false, afrag[kb], false, b0, (short)0, acc0, false, false);
        acc1 = __builtin_amdgcn_wmma_f32_16x16x4_f32(false, afrag[kb], false, b1, (short)0, acc1, false, false);
        acc2 = __builtin_amdgcn_wmma_f32_16x16x4_f32(false, afrag[kb], false, b2, (short)0, acc2, false, false);
        acc3 = __builtin_amdgcn_wmma_f32_16x16x4_f32(false, afrag[kb], false, b3, (short)0, acc3, false, false);
    }

    // D layout: lanes 0-15 hold M = r, lanes 16-31 hold M = r+8, col = lane&15 (+16*ntile)
    int mrow = row_base + (half << 3);
    for (int r = 0; r < 8; ++r) {
        float* o = Out + (long long)(mrow + r) * NF + m;
        o[0]  = acc0[r];
        o[16] = acc1[r];
        o[32] = acc2[r];
        o[48] = acc3[r];
    }
}

// ---------------- bias init + normalized scatter-add ----------------
__global__ void k_init_bias(float* __restrict__ agg, const float* __restrict__ b, int n) {
    int t = blockIdx.x * blockDim.x + threadIdx.x;
    if (t < n * NF) agg[t] = b[t & (NF - 1)];
}

// One wave per item (edge or self-loop); lane handles 2 features.
// Item index forced wave-uniform so src/dst/dinv become scalar loads.
__global__ __launch_bounds__(256) void k_scatter(const float* __restrict__ h,
                                                 const int* __restrict__ src,
                                                 const int* __restrict__ dst,
                                                 const float* __restrict__ dinv,
                                                 float* __restrict__ agg,
                                                 int e, int n) {
    int wid  = __builtin_amdgcn_readfirstlane(threadIdx.x >> 5);
    int lane = threadIdx.x & 31;
    int item = blockIdx.x * 8 + wid;  // 8 waves per 256-thread block
    if (item >= e + n) return;
    int s, d;
    float nrm;
    if (item < e) {
        s = src[item];
        d = dst[item];
        nrm = dinv[s] * dinv[d];
    } else {
        s = d = item - e;
        float di = dinv[s];
        nrm = di * di;
    }
    int f = lane << 1;
    v2f hv = *(const v2f*)(h + (long long)s * NF + f);
    float* ap = agg + (long long)d * NF + f;
    unsafeAtomicAdd(ap,     hv.x * nrm);
    unsafeAtomicAdd(ap + 1, hv.y * nrm);
}

// ---------------- pooling + head ----------------
__global__ void k_zero_pool(float* __restrict__ pooled, float* __restrict__ counts, int g) {
    int t = blockIdx.x * blockDim.x + threadIdx.x;
    if (t < g * NF) pooled[t] = 0.0f;
    else if (t < g * NF + g) counts[t - g * NF] = 0.0f;
}

// One wave per node; lane handles 2 features; batch[] becomes a scalar load.
__global__ __launch_bounds__(256) void k_pool(const float* __restrict__ agg,
                                              const int* __restrict__ batch,
                                              float* __restrict__ pooled,
                                              float* __restrict__ counts, int n) {
    int wid  = __builtin_amdgcn_readfirstlane(threadIdx.x >> 5);
    int lane = threadIdx.x & 31;
    int i = blockIdx.x * 8 + wid;
    if (i >= n) return;
    int g = batch[i];
    int f = lane << 1;
    v2f v = *(const v2f*)(agg + (long long)i * NF + f);
    float* pp = pooled + g * NF + f;
    unsafeAtomicAdd(pp,     fmaxf(v.x, 0.0f));  // relu of layer-2 preactivation
    unsafeAtomicAdd(pp + 1, fmaxf(v.y, 0.0f));
    if (lane == 0) unsafeAtomicAdd(&counts[g], 1.0f);
}

__global__ void k_head(const float* __restrict__ pooled,
                       const float* __restrict__ counts,
                       const float* __restrict__ Wfc,
                       const float* __restrict__ bfc,
                       float* __restrict__ out, int g) {
    int i = blockIdx.x * blockDim.x + threadIdx.x;
    if (i >= g) return;
    float c = fmaxf(counts[i], 1.0f);
    float acc = bfc[0];
    float inv = 1.0f / c;
    for (int f = 0; f < NF; ++f) acc += pooled[i * NF + f] * inv * Wfc[f];
    out[i] = 1.0f / (1.0f + __expf(-acc));
}

// ---------------- driver ----------------
extern "C" void kernel_launch(void* const* d_in, const int* in_sizes, int n_in,
                              void* d_out, int out_size, void* d_ws, size_t ws_size,
                              hipStream_t stream) {
    const float* x     = (const float*)d_in[0];
    const int*   ei    = (const int*)d_in[1];
    const int*   batch = (const int*)d_in[2];
    const float* W1    = (const float*)d_in[3];
    const float* b1    = (const float*)d_in[4];
    const float* W2    = (const float*)d_in[5];
    const float* b2    = (const float*)d_in[6];
    const float* Wfc   = (const float*)d_in[7];
    const float* bfc   = (const float*)d_in[8];
    float* out = (float*)d_out;

    int N = in_sizes[0] / NF;     // 100000
    int E = in_sizes[1] / 2;      // 1600000
    int G = out_size;             // 256

    const int* esrc = ei;
    const int* edst = ei + E;

    float* ws     = (float*)d_ws;
    float* bufH   = ws;                          // N*64 : GEMM output h = A @ W
    float* bufG   = bufH + (size_t)N * NF;       // N*64 : aggregated preactivation
    float* deg    = bufG + (size_t)N * NF;       // N    : degree -> dinv (in place)
    float* pooled = deg + N;                     // G*64
    float* counts = pooled + (size_t)G * NF;     // G

    dim3 blk(256);
    auto cdiv = [](long long a, long long b) { return (unsigned)((a + b - 1) / b); };

    // normalization factors
    k_init_deg<<<cdiv(N, 256), blk, 0, stream>>>(deg, N);
    k_deg_accum<<<cdiv(E, 256), blk, 0, stream>>>(edst, deg, E);
    k_dinv<<<cdiv(N, 256), blk, 0, stream>>>(deg, N);

    unsigned gemm_grid     = cdiv(N, 64);          // 4 waves x 16 rows per block
    unsigned scatter_grid  = cdiv(E + N, 8);       // 1 wave per edge/self-loop
    unsigned pool_grid     = cdiv(N, 8);           // 1 wave per node
    unsigned nodefeat_grid = cdiv((long long)N * NF, 256);

    // layer 1: h1 = X @ W1 ; agg1 = scatter(h1) + b1
    k_gemm64<<<gemm_grid, 128, 0, stream>>>(x, W1, bufH, N, 0);
    k_init_bias<<<nodefeat_grid, blk, 0, stream>>>(bufG, b1, N);
    k_scatter<<<scatter_grid, blk, 0, stream>>>(bufH, esrc, edst, deg, bufG, E, N);

    // layer 2: h2 = relu(agg1) @ W2 ; agg2 = scatter(h2) + b2
    k_gemm64<<<gemm_grid, 128, 0, stream>>>(bufG, W2, bufH, N, 1);
    k_init_bias<<<nodefeat_grid, blk, 0, stream>>>(bufG, b2, N);
    k_scatter<<<scatter_grid, blk, 0, stream>>>(bufH, esrc, edst, deg, bufG, E, N);

    // mean pool of relu(agg2) + sigmoid head
    k_zero_pool<<<cdiv((long long)G * (NF + 1), 256), blk, 0, stream>>>(pooled, counts, G);
    k_pool<<<pool_grid, blk, 0, stream>>>(bufG, batch, pooled, counts, N);
    k_head<<<cdiv(G, 256), blk, 0, stream>>>(pooled, counts, Wfc, bfc, out, G);
}